// upper_module_61572651156040
// MI455X (gfx1250) — compile-verified
//
#include <hip/hip_runtime.h>

typedef __attribute__((ext_vector_type(2))) float v2f;
typedef __attribute__((ext_vector_type(4))) float v4f;
typedef __attribute__((ext_vector_type(8))) float v8f;

#define B_   256
#define P_   90
#define H_   64
#define W_   64
#define FXY  64        // 8*8 pooled features per pair
#define CV   1024
#define MROWS (B_ * P_)   // 23040, divisible by 16

// ---------------------------------------------------------------------------
// Kernel 1: adaptive 8x8 max pool + attention score.
// One 64-thread block per (b,p) tile; thread t = output cell (wy,wx).
// ---------------------------------------------------------------------------
__global__ __launch_bounds__(64) void pool_score_kernel(
    const float* __restrict__ features, const float* __restrict__ w_att,
    const float* __restrict__ b_att, float* __restrict__ pooled,
    float* __restrict__ scores)
{
    const int bp = blockIdx.x;            // 0 .. B*P-1
    const int t  = threadIdx.x;           // 0 .. 63
    const int wy = t >> 3, wx = t & 7;
    const float* tile = features + (size_t)bp * (H_ * W_);

    float m = -INFINITY;
#pragma unroll
    for (int r = 0; r < 8; ++r) {
        const float* rowp = tile + (size_t)(wy * 8 + r) * W_ + wx * 8;
        v4f a = *(const v4f*)(rowp);
        v4f b = *(const v4f*)(rowp + 4);
        float ra = fmaxf(fmaxf(a.x, a.y), fmaxf(a.z, a.w));
        float rb = fmaxf(fmaxf(b.x, b.y), fmaxf(b.z, b.w));
        m = fmaxf(m, fmaxf(ra, rb));
    }
    pooled[(size_t)bp * FXY + t] = m;

    __shared__ float red[64];
    red[t] = m * w_att[t];
    __syncthreads();
#pragma unroll
    for (int s = 32; s > 0; s >>= 1) {
        if (t < s) red[t] += red[t + s];
        __syncthreads();
    }
    if (t == 0) scores[bp] = red[0] + b_att[0];
}

// ---------------------------------------------------------------------------
// Kernel 2: per-batch top-k by rank; coef = score if selected else 0.
// Rank = #{q : s_q > s_p || (s_q == s_p && q < p)} matches top_k tiebreak.
// ---------------------------------------------------------------------------
__global__ __launch_bounds__(128) void topk_coef_kernel(
    const float* __restrict__ scores, const int* __restrict__ kptr,
    float* __restrict__ coef)
{
    const int b = blockIdx.x;
    const int t = threadIdx.x;
    const int k = *kptr;

    __shared__ float s[P_];
    if (t < P_) s[t] = scores[b * P_ + t];
    __syncthreads();

    if (t < P_) {
        const float my = s[t];
        int rank = 0;
        for (int q = 0; q < P_; ++q) {
            const float sq = s[q];
            rank += ((sq > my) || (sq == my && q < t)) ? 1 : 0;
        }
        coef[b * P_ + t] = (rank < k) ? my : 0.0f;
    }
}

// ---------------------------------------------------------------------------
// Kernel 3: out[M,CV] = (pooled * coef)[M,64] @ w_fc[64,CV] + b_fc, with
// V_WMMA_F32_16X16X4_F32. One wave per 16x16 output tile; 8 waves/block
// sharing one M-row of tiles (A reuse). K=64 -> 16 WMMA steps.
//
// A 16x4 f32 layout: lanes 0-15 hold M=lane with K = k0,k0+1 in the two
// VGPRs; lanes 16-31 hold the same rows with K = k0+2,k0+3.
// B 4x16 layout mirrors: lanes 0-15 hold N=lane, K = k0,k0+1; lanes 16-31
// hold N=lane-16, K = k0+2,k0+3.
// D: VGPR r, lanes 0-15 -> row m0+r; lanes 16-31 -> row m0+8+r.
// ---------------------------------------------------------------------------
__global__ __launch_bounds__(256) void wmma_fc_kernel(
    const float* __restrict__ pooled, const float* __restrict__ coef,
    const float* __restrict__ w_fc, const float* __restrict__ b_fc,
    float* __restrict__ out)
{
    const int lane = threadIdx.x & 31;
    const int wave = threadIdx.x >> 5;
    const int NT   = CV / 16;                       // 64 N-tiles
    const int tile = blockIdx.x * 8 + wave;         // global tile id
    const int mt   = tile / NT;
    const int nt   = tile % NT;
    const int m0   = mt * 16;
    const int n0   = nt * 16;

    const int half = lane >> 4;       // 0: K pair (k0,k0+1); 1: (k0+2,k0+3)
    const int l    = lane & 15;
    const int koff = half * 2;

    const int   row  = m0 + l;                      // A row fed by this lane
    const float c    = coef[row];                   // 0 for non-selected pairs
    const float* arow = pooled + (size_t)row * FXY;

    v8f acc = {};
#pragma unroll
    for (int kk = 0; kk < 16; ++kk) {
        const int k0 = kk * 4;
        v2f a;
        a.x = arow[k0 + koff]     * c;
        a.y = arow[k0 + koff + 1] * c;
        v2f bm;
        bm.x = w_fc[(size_t)(k0 + koff)     * CV + n0 + l];
        bm.y = w_fc[(size_t)(k0 + koff + 1) * CV + n0 + l];
        // 8 args: (neg_a, A, neg_b, B, c_mod, C, reuse_a, reuse_b)
        acc = __builtin_amdgcn_wmma_f32_16x16x4_f32(
            false, a, false, bm, (short)0, acc, false, false);
    }

    const int   ncol  = n0 + l;
    const float bias  = b_fc[ncol];
    const int   mbase = m0 + half * 8;
#pragma unroll
    for (int r = 0; r < 8; ++r) {
        out[(size_t)(mbase + r) * CV + ncol] = acc[r] + bias;
    }
}

// ---------------------------------------------------------------------------
extern "C" void kernel_launch(void* const* d_in, const int* in_sizes, int n_in,
                              void* d_out, int out_size, void* d_ws, size_t ws_size,
                              hipStream_t stream)
{
    (void)in_sizes; (void)n_in; (void)out_size; (void)ws_size;

    const float* features = (const float*)d_in[0];   // [B,P,H,W]
    const float* w_att    = (const float*)d_in[1];   // [64,1]
    const float* b_att    = (const float*)d_in[2];   // [1]
    const float* w_fc     = (const float*)d_in[3];   // [64,1024]
    const float* b_fc     = (const float*)d_in[4];   // [1024]
    const int*   kptr     = (const int*)d_in[5];     // [1]
    float*       out      = (float*)d_out;           // [B,P,1024]

    // Workspace layout (floats): pooled[MROWS*64] | scores[MROWS] | coef[MROWS]
    float* pooled = (float*)d_ws;
    float* scores = pooled + (size_t)MROWS * FXY;
    float* coefs  = scores + MROWS;

    pool_score_kernel<<<MROWS, 64, 0, stream>>>(features, w_att, b_att,
                                                pooled, scores);
    topk_coef_kernel<<<B_, 128, 0, stream>>>(scores, kptr, coefs);

    const int n_tiles = (MROWS / 16) * (CV / 16);    // 92160
    wmma_fc_kernel<<<n_tiles / 8, 256, 0, stream>>>(pooled, coefs, w_fc,
                                                    b_fc, out);
}